// SimOTA_24910810317319
// MI455X (gfx1250) — compile-verified
//
#include <hip/hip_runtime.h>
#include <hip/hip_bf16.h>
#include <math.h>

typedef float v2f __attribute__((ext_vector_type(2)));
typedef float v8f __attribute__((ext_vector_type(8)));

#define TPI 20
#define NC 80
#define PSTR 85   // 4 box + 1 obj + 80 cls
#define TOPK 10

#define LOG2E 1.4426950408889634f
#define LN2   0.6931471805599453f

// Raw gfx1250 hardware transcendentals (no denorm guards / fixup chains).
__device__ __forceinline__ float rcp_hw(float x)  { return __builtin_amdgcn_rcpf(x); }
__device__ __forceinline__ float exp2_hw(float x) { return __builtin_amdgcn_exp2f(x); }  // v_exp_f32 (2^x)
__device__ __forceinline__ float log2_hw(float x) { return __builtin_amdgcn_logf(x); }   // v_log_f32 (log2 x)
__device__ __forceinline__ float sqrt_hw(float x) { return __builtin_amdgcn_sqrtf(x); }  // v_sqrt_f32

__device__ __forceinline__ float sig_hw(float x) {
    return rcp_hw(1.0f + exp2_hw(-x * LOG2E));
}
__device__ __forceinline__ float bcef(float z, float y) {
    // max(z,0) - z*y + log1p(exp(-|z|)) via hardware exp2/log2
    return fmaxf(z, 0.0f) - z * y + LN2 * log2_hw(1.0f + exp2_hw(-fabsf(z) * LOG2E));
}

// ---------------------------------------------------------------- K0: zero
__global__ void k_zero(unsigned int* mm, float* acc, int n) {
    int g = blockIdx.x * blockDim.x + threadIdx.x;
    if (g < n) mm[g] = 0u;
    if (g < 8) acc[g] = 0.0f;
}

// ---------------------------------------------------------------- K1: cost/iou/fg
// 256 threads = 8 waves; each wave owns one 16-anchor tile of one image.
// cls_cost = -(S[a] + (Y @ D^T)[t,a]); the matmul runs on V_WMMA_F32_16X16X4_F32.
__global__ __launch_bounds__(256) void k_cost(
    const float* __restrict__ preds, const float* __restrict__ grid,
    const float* __restrict__ stridem, const float* __restrict__ targets,
    float* __restrict__ costw, float* __restrict__ iouw,
    unsigned char* __restrict__ fgw, int A)
{
    __shared__ float4 tb4[TPI];
    __shared__ int    scid[TPI];
    __shared__ int    svld[TPI];

    const int b = blockIdx.y;
    if (threadIdx.x < TPI) {
        const float* t = targets + (size_t)(b * TPI + threadIdx.x) * 6;
        svld[threadIdx.x] = ((int)t[0] == b) ? 1 : 0;
        scid[threadIdx.x] = (int)t[1];
        tb4[threadIdx.x] = make_float4(t[2], t[3], t[4], t[5]);
    }
    __syncthreads();

    const int tiles = A >> 4;
    const int tile = blockIdx.x * 8 + (threadIdx.x >> 5);
    if (tile >= tiles) return;

    const int lane = threadIdx.x & 31;
    const int n  = lane & 15;   // anchor within tile (== M row index of A-frag)
    const int kh = lane >> 4;   // K-half per f32 WMMA fragment layout
    const int a  = tile * 16 + n;

    const float* p = preds + ((size_t)b * A + a) * PSTR;
    const float px1 = p[0], py1 = p[1], px2 = p[2], py2 = p[3];
    const float sobj = sig_hw(p[4]);
    const float sstr = stridem[a];
    const float xc = (grid[2 * a]     + 0.5f) * sstr;
    const float yc = (grid[2 * a + 1] + 0.5f) * sstr;

    // Hoisted one-hot class ids (Y row is one-hot * valid): no LDS in the loop.
    const int m2 = 16 + n;
    const int cidm1 = svld[n] ? scid[n] : -1;
    const int cidm2 = (m2 < TPI && svld[m2]) ? scid[m2] : -1;

    v8f acc0 = {0, 0, 0, 0, 0, 0, 0, 0};  // targets 0..15
    v8f acc1 = {0, 0, 0, 0, 0, 0, 0, 0};  // targets 16..19 (padded to 32)
    float S2 = 0.0f;                       // sum of log2(1-q); converted by *LN2 once

    #pragma unroll
    for (int kk = 0; kk < NC / 4; ++kk) {
        // This lane supplies K = {c0, c0+1} of the 4-wide K chunk.
        const int c0 = kk * 4 + 2 * kh;
        const float z0 = p[5 + c0], z1 = p[6 + c0];
        float q0 = sqrt_hw(sobj * sig_hw(z0));
        float q1 = sqrt_hw(sobj * sig_hw(z1));
        q0 = fminf(fmaxf(q0, 1e-7f), 1.0f - 1e-7f);
        q1 = fminf(fmaxf(q1, 1e-7f), 1.0f - 1e-7f);
        const float l2m0 = log2_hw(1.0f - q0), l2m1 = log2_hw(1.0f - q1);
        const float d0 = (log2_hw(q0) - l2m0) * LN2;
        const float d1 = (log2_hw(q1) - l2m1) * LN2;
        S2 += l2m0 + l2m1;

        v2f bf; bf.x = d0; bf.y = d1;                 // B frag: D[anchor n][K]
        v2f a0;                                        // A frag: Y[target n][K]
        a0.x = (cidm1 == c0)     ? 1.0f : 0.0f;
        a0.y = (cidm1 == c0 + 1) ? 1.0f : 0.0f;
        v2f a1;                                        // A frag tile 1 (rows >=20 zero)
        a1.x = (cidm2 == c0)     ? 1.0f : 0.0f;
        a1.y = (cidm2 == c0 + 1) ? 1.0f : 0.0f;

        acc0 = __builtin_amdgcn_wmma_f32_16x16x4_f32(false, a0, false, bf, (short)0, acc0, false, false);
        acc1 = __builtin_amdgcn_wmma_f32_16x16x4_f32(false, a1, false, bf, (short)0, acc1, false, false);
    }
    // Full S[a] = ln-domain sum over all 80 classes (each lane holds half the K space).
    float S = S2 * LN2;
    S += __shfl_xor(S, 16, 32);

    unsigned fgacc = 0;
    auto doPair = [&](int t, float ydot) {
        const float4 tbv = tb4[t];
        const float tx1 = tbv.x, ty1 = tbv.y, tx2 = tbv.z, ty2 = tbv.w;
        const int tvld = svld[t];
        // IoU(tgt, pred)
        const float aa = (tx2 - tx1) * (ty2 - ty1);
        const float ab = (px2 - px1) * (py2 - py1);
        const float lx = fmaxf(tx1, px1), ly = fmaxf(ty1, py1);
        const float rx = fminf(tx2, px2), ry = fminf(ty2, py2);
        const float iw = fmaxf(rx - lx, 0.0f), ih = fmaxf(ry - ly, 0.0f);
        const float inter = iw * ih;
        const float iou = inter * rcp_hw(aa + ab - inter + 1e-8f);
        // in_boxes / in_centers
        const float dmin = fminf(fminf(xc - tx1, yc - ty1), fminf(tx2 - xc, ty2 - yc));
        const int inb = (dmin > 0.0f) & tvld;
        const float cxt = 0.5f * (tx1 + tx2), cyt = 0.5f * (ty1 + ty2);
        const int inc = (fmaxf(fabsf(xc - cxt), fabsf(yc - cyt)) < 2.5f * sstr) & tvld;
        fgacc |= (unsigned)(inb | inc);

        float cost = -(S + ydot) - (3.0f * LN2) * log2_hw(iou + 1e-8f);
        cost += (inb & inc) ? 0.0f : 100000.0f;
        cost += tvld ? 0.0f : 1000000000.0f;
        // NOTE: the 1e6*(!fg) term is per-anchor-constant; added in k_select.
        const size_t o = (size_t)(b * TPI + t) * A + a;
        costw[o] = cost;
        iouw[o]  = iou;
    };

    // C/D layout: VGPR r -> M = r + 8*kh, N = lane&15
    #pragma unroll
    for (int r = 0; r < 8; ++r) { doPair(r + 8 * kh, acc0[r]); }
    if (kh == 0) {
        #pragma unroll
        for (int r = 0; r < 4; ++r) { doPair(16 + r, acc1[r]); }
    }
    const unsigned fgo = fgacc | __shfl_xor(fgacc, 16, 32);
    if (kh == 0) fgw[(size_t)b * A + a] = (unsigned char)(fgo & 1u);
}

// ---------------------------------------------------------------- K2: dynamic top-k select
// One wave per (image, target). Lexicographic (cost, index) matches lax.top_k tie-breaking.
__global__ __launch_bounds__(32) void k_select(
    const float* __restrict__ costw, const float* __restrict__ iouw,
    const unsigned char* __restrict__ fgw, const float* __restrict__ targets,
    unsigned int* __restrict__ mmw, int A)
{
    const int bt = blockIdx.x;
    const int b = bt / TPI, t = bt % TPI;
    if ((int)targets[(size_t)bt * 6] != b) return;  // sel requires valid

    const int lane = threadIdx.x;
    const float* crow = costw + (size_t)bt * A;
    const float* irow = iouw  + (size_t)bt * A;
    const unsigned char* fg = fgw + (size_t)b * A;

    float c10[TOPK]; int i10[TOPK]; float v10[TOPK];
    #pragma unroll
    for (int i = 0; i < TOPK; ++i) { c10[i] = 3.4e38f; i10[i] = 0x7fffffff; v10[i] = -1.0f; }

    for (int a = lane; a < A; a += 32) {
        const int f = fg[a];
        const float c = crow[a] + (f ? 0.0f : 1000000.0f);
        const float iv = f ? irow[a] : 0.0f;
        // keep 10 smallest (c, a)
        if (c < c10[TOPK - 1] || (c == c10[TOPK - 1] && a < i10[TOPK - 1])) {
            int j = TOPK - 1;
            while (j > 0 && (c < c10[j - 1] || (c == c10[j - 1] && a < i10[j - 1]))) {
                c10[j] = c10[j - 1]; i10[j] = i10[j - 1]; --j;
            }
            c10[j] = c; i10[j] = a;
        }
        // keep 10 largest masked iou
        if (iv > v10[TOPK - 1]) {
            int j = TOPK - 1;
            while (j > 0 && iv > v10[j - 1]) { v10[j] = v10[j - 1]; --j; }
            v10[j] = iv;
        }
    }

    __shared__ float sc[32 * TOPK];
    __shared__ int   si[32 * TOPK];
    __shared__ float sv[32 * TOPK];
    #pragma unroll
    for (int i = 0; i < TOPK; ++i) {
        sc[lane * TOPK + i] = c10[i];
        si[lane * TOPK + i] = i10[i];
        sv[lane * TOPK + i] = v10[i];
    }
    __syncthreads();

    if (lane == 0) {
        float fc[TOPK]; int fi[TOPK]; float fv[TOPK];
        #pragma unroll
        for (int i = 0; i < TOPK; ++i) { fc[i] = 3.4e38f; fi[i] = 0x7fffffff; fv[i] = -1.0f; }
        for (int i = 0; i < 32 * TOPK; ++i) {
            const float c = sc[i]; const int ix = si[i]; const float iv = sv[i];
            if (c < fc[TOPK - 1] || (c == fc[TOPK - 1] && ix < fi[TOPK - 1])) {
                int j = TOPK - 1;
                while (j > 0 && (c < fc[j - 1] || (c == fc[j - 1] && ix < fi[j - 1]))) {
                    fc[j] = fc[j - 1]; fi[j] = fi[j - 1]; --j;
                }
                fc[j] = c; fi[j] = ix;
            }
            if (iv > fv[TOPK - 1]) {
                int j = TOPK - 1;
                while (j > 0 && iv > fv[j - 1]) { fv[j] = fv[j - 1]; --j; }
                fv[j] = iv;
            }
        }
        float s = 0.0f;
        #pragma unroll
        for (int i = 0; i < TOPK; ++i) s += fmaxf(fv[i], 0.0f);
        int dynk = (int)s;                 // trunc toward zero, values >= 0
        dynk = dynk < 1 ? 1 : (dynk > TOPK ? TOPK : dynk);
        for (int j = 0; j < dynk; ++j)
            atomicOr(&mmw[(size_t)b * A + fi[j]], 1u << t);
    }
}

// ---------------------------------------------------------------- K3: resolve + fused loss
__global__ __launch_bounds__(256) void k_loss(
    const float* __restrict__ preds, const float* __restrict__ targets,
    const float* __restrict__ costw, const float* __restrict__ iouw,
    const unsigned char* __restrict__ fgw, const unsigned int* __restrict__ mmw,
    float* __restrict__ acc, int A, int B)
{
    __shared__ float sacc[4];
    if (threadIdx.x < 4) sacc[threadIdx.x] = 0.0f;
    __syncthreads();

    const int g = blockIdx.x * blockDim.x + threadIdx.x;
    float lcls = 0.0f, lbox = 0.0f, lobj = 0.0f, npos = 0.0f;

    if (g < B * A) {
        const int b = g / A, a = g % A;
        unsigned mask = fgw[g] ? mmw[g] : 0u;     // mm & fg
        if (__popc(mask) > 1) {                   // multi-match -> argmin over t of cost
            float best = costw[(size_t)(b * TPI) * A + a];
            int bt = 0;
            for (int t = 1; t < TPI; ++t) {
                const float c = costw[(size_t)(b * TPI + t) * A + a];
                if (c < best) { best = c; bt = t; }
            }
            mask = 1u << bt;
        }
        const float* p = preds + (size_t)g * PSTR;
        const float zo = p[4];
        const int matched = (mask != 0u);
        lcls = bcef(zo, matched ? 1.0f : 0.0f);
        if (matched) {
            npos = 1.0f;
            const int at = __ffs(mask) - 1;       // argmax of bool -> first set t
            const float piou = iouw[(size_t)(b * TPI + at) * A + a];
            const float* tg = targets + (size_t)(b * TPI + at) * 6;
            const int cc = (int)tg[1];
            const float bx1 = tg[2], by1 = tg[3], bx2 = tg[4], by2 = tg[5];
            const float ax1 = p[0], ay1 = p[1], ax2 = p[2], ay2 = p[3];
            // CIoU(pred, target)
            const float lx = fmaxf(ax1, bx1), ly = fmaxf(ay1, by1);
            const float rx = fminf(ax2, bx2), ry = fminf(ay2, by2);
            const float iw = fmaxf(rx - lx, 0.0f), ih = fmaxf(ry - ly, 0.0f);
            const float inter = iw * ih;
            const float wa = ax2 - ax1, ha = ay2 - ay1;
            const float wb = bx2 - bx1, hb = by2 - by1;
            const float uni = wa * ha + wb * hb - inter + 1e-7f;
            const float iou = inter / uni;
            const float cw = fmaxf(ax2, bx2) - fminf(ax1, bx1);
            const float ch = fmaxf(ay2, by2) - fminf(ay1, by1);
            const float c2 = cw * cw + ch * ch + 1e-7f;
            const float dx = ax1 + ax2 - bx1 - bx2;
            const float dy = ay1 + ay2 - by1 - by2;
            const float rho2 = (dx * dx + dy * dy) * 0.25f;
            const float dat = atanf(wb / (hb + 1e-7f)) - atanf(wa / (ha + 1e-7f));
            const float v = (4.0f / 9.8696044010893586f) * dat * dat;
            const float alpha = v / (1.0f - iou + v + 1e-7f);
            lbox = 1.0f - (iou - rho2 / c2 - alpha * v);
            // smoothed class BCE over 80 logits
            float s = 0.0f;
            for (int c = 0; c < NC; ++c) {
                const float y = (c == cc && piou > 0.0f) ? (1.0f - 0.05f) : (0.05f / (NC - 1));
                s += bcef(p[5 + c], y);
            }
            lobj = s;
        }
    }
    atomicAdd(&sacc[0], lcls);
    atomicAdd(&sacc[1], lbox);
    atomicAdd(&sacc[2], lobj);
    atomicAdd(&sacc[3], npos);
    __syncthreads();
    if (threadIdx.x < 4) atomicAdd(&acc[threadIdx.x], sacc[threadIdx.x]);
}

// ---------------------------------------------------------------- K4: finalize
__global__ void k_final(const float* __restrict__ acc, float* __restrict__ out, int A, int B) {
    if (threadIdx.x == 0 && blockIdx.x == 0) {
        const float npos = fmaxf(acc[3], 1.0f);
        const float lbox = acc[1] / npos;
        const float lobj = acc[2] / (npos * (float)NC);
        const float lcls = acc[0] / (float)(B * A);
        out[0] = lbox + lobj + lcls;
        out[1] = lbox;
        out[2] = lobj;
        out[3] = lcls;
    }
}

extern "C" void kernel_launch(void* const* d_in, const int* in_sizes, int n_in,
                              void* d_out, int out_size, void* d_ws, size_t ws_size,
                              hipStream_t stream) {
    const float* preds   = (const float*)d_in[0];
    const float* grid    = (const float*)d_in[1];
    const float* stridem = (const float*)d_in[2];
    const float* targets = (const float*)d_in[3];
    const int A = in_sizes[2];                 // 33600 (divisible by 16)
    const int B = in_sizes[3] / (TPI * 6);     // 16

    // workspace layout: cost[B*T*A] f32 | iou[B*T*A] f32 | mm[B*A] u32 | acc[8] f32 | fg[B*A] u8
    const size_t nTA = (size_t)B * TPI * A;
    float*         costw = (float*)d_ws;
    float*         iouw  = costw + nTA;
    unsigned int*  mmw   = (unsigned int*)(iouw + nTA);
    float*         acc   = (float*)(mmw + (size_t)B * A);
    unsigned char* fgw   = (unsigned char*)(acc + 8);

    const int nz = B * A;
    hipLaunchKernelGGL(k_zero, dim3((nz + 255) / 256), dim3(256), 0, stream, mmw, acc, nz);

    const int tiles = A / 16;
    dim3 g1((tiles + 7) / 8, B);
    hipLaunchKernelGGL(k_cost, g1, dim3(256), 0, stream,
                       preds, grid, stridem, targets, costw, iouw, fgw, A);

    hipLaunchKernelGGL(k_select, dim3(B * TPI), dim3(32), 0, stream,
                       costw, iouw, fgw, targets, mmw, A);

    hipLaunchKernelGGL(k_loss, dim3((nz + 255) / 256), dim3(256), 0, stream,
                       preds, targets, costw, iouw, fgw, mmw, acc, A, B);

    hipLaunchKernelGGL(k_final, dim3(1), dim3(1), 0, stream, acc, (float*)d_out, A, B);
}